// TransformerEncoderLayer_69630009803109
// MI455X (gfx1250) — compile-verified
//
#include <hip/hip_runtime.h>
#include <hip/hip_bf16.h>
#include <stdint.h>

// ---------------- shapes (fixed by reference) ----------------
#define BB 4
#define LL 512
#define DD 256
#define HH 8
#define DHD 32
#define CC 64
#define SS 16
#define FF 1024

typedef __attribute__((ext_vector_type(2))) float v2f;
typedef __attribute__((ext_vector_type(8))) float v8f;

// ---------------- WMMA f32 16x16x4 wrapper (codegen-confirmed) ----------------
static __device__ inline v8f wmma_f32_4(v2f a, v2f b, v8f c) {
#if __has_builtin(__builtin_amdgcn_wmma_f32_16x16x4_f32)
  return __builtin_amdgcn_wmma_f32_16x16x4_f32(false, a, false, b, (short)0, c,
                                               false, false);
#else
#warning "wmma_f32_16x16x4_f32 builtin not available - scalar placeholder"
  c[0] += a[0] * b[0] + a[1] * b[1];
  return c;
#endif
}

static __device__ inline float wred_max(float v) {
#pragma unroll
  for (int i = 16; i > 0; i >>= 1) v = fmaxf(v, __shfl_xor(v, i, 32));
  return v;
}
static __device__ inline float wred_sum(float v) {
#pragma unroll
  for (int i = 16; i > 0; i >>= 1) v += __shfl_xor(v, i, 32);
  return v;
}

// e3nn real SH l=0..3, integral norm, with (x,y,z)->(Y,Z,X) axis map
static __device__ inline void sph16(float dx, float dy, float dz, float* sh) {
  float n2 = dx * dx + dy * dy + dz * dz;
  float inv = 1.0f / fmaxf(sqrtf(n2), 1e-12f);
  float ux = dx * inv, uy = dy * inv, uz = dz * inv;
  float X = uz, Y = ux, Z = uy;
  float XX = X * X, YY = Y * Y, ZZ = Z * Z;
  sh[0] = 0.28209479177387814f;
  sh[1] = 0.4886025119029199f * Y;
  sh[2] = 0.4886025119029199f * Z;
  sh[3] = 0.4886025119029199f * X;
  sh[4] = 1.0925484305920792f * X * Y;
  sh[5] = 1.0925484305920792f * Y * Z;
  sh[6] = 0.31539156525252005f * (3.0f * ZZ - 1.0f);
  sh[7] = 1.0925484305920792f * X * Z;
  sh[8] = 0.5462742152960396f * (XX - YY);
  sh[9] = 0.5900435899266435f * Y * (3.0f * XX - YY);
  sh[10] = 2.890611442640554f * X * Y * Z;
  sh[11] = 0.4570457994644658f * Y * (5.0f * ZZ - 1.0f);
  sh[12] = 0.3731763325901154f * Z * (5.0f * ZZ - 3.0f);
  sh[13] = 0.4570457994644658f * X * (5.0f * ZZ - 1.0f);
  sh[14] = 1.445305721320277f * Z * (XX - YY);
  sh[15] = 0.5900435899266435f * X * (XX - 3.0f * YY);
}

// ============ kernel 1: fold q into rel/dir weights =============
// grid = B*L blocks, 256 threads
__global__ void kprep(const float* __restrict__ src,
                      const float* __restrict__ relW,
                      const float* __restrict__ relb,
                      const float* __restrict__ dirW,
                      const float* __restrict__ dirb,
                      float* __restrict__ qWr, float* __restrict__ qWd,
                      float* __restrict__ qb) {
  int bl = blockIdx.x;
  int t = threadIdx.x;
  __shared__ float q[DD];
  q[t] = src[(size_t)bl * DD + t];
  __syncthreads();
  for (int i = t; i < HH * CC; i += 256) {
    int h = i >> 6, c = i & 63;
    const float* w = relW + (size_t)c * DD + h * DHD;
    const float* qq = q + h * DHD;
    float s = 0.f;
#pragma unroll
    for (int d = 0; d < DHD; ++d) s += qq[d] * w[d];
    qWr[(size_t)bl * (HH * CC) + i] = s;
  }
  if (t < HH * SS) {
    int h = t >> 4, s16 = t & 15;
    const float* w = dirW + (size_t)s16 * DD + h * DHD;
    const float* qq = q + h * DHD;
    float s = 0.f;
#pragma unroll
    for (int d = 0; d < DHD; ++d) s += qq[d] * w[d];
    qWd[(size_t)bl * (HH * SS) + t] = s;
  }
  if (t < HH) {
    const float* qq = q + t * DHD;
    float s = 0.f;
#pragma unroll
    for (int d = 0; d < DHD; ++d)
      s += qq[d] * (relb[t * DHD + d] + dirb[t * DHD + d]);
    qb[(size_t)bl * HH + t] = s;
  }
}

// ============ kernel 2: attention + softmax + AV + LN1 =============
// grid = B*L blocks, 512 threads (16 waves)
// rel_features slabs double-buffered in LDS via async-to-LDS DMA.
#define RFP 68   // padded rel-feature row stride (floats)
#define WRP 68   // padded qWr row stride
#define WDP 20   // padded qWd row stride
#define SCP 520  // padded score row stride
__global__ void __launch_bounds__(512) kattn(
    const float* __restrict__ src, const float* __restrict__ relF,
    const float* __restrict__ relD, const float* __restrict__ qWr,
    const float* __restrict__ qWd, const float* __restrict__ qb,
    const float* __restrict__ g1, const float* __restrict__ bb1,
    float* __restrict__ x1out) {
  __shared__ float qWrS[HH * WRP];
  __shared__ float qWdS[HH * WDP];
  __shared__ float qbS[HH];
  __shared__ float qlS[DD];
  __shared__ float rfS[2][64 * RFP];  // double-buffered 16KB slabs
  __shared__ float sphS[64 * SS];
  __shared__ float scoresS[HH * SCP];
  __shared__ float redS[DD];

  const int bl = blockIdx.x;
  const int b = bl >> 9;
  const int t = threadIdx.x;
  const int wave = t >> 5;
  const int lane = t & 31;
  const float scale = 0.17677669529663687f;  // 1/sqrt(32)

  if (t < HH * CC) qWrS[(t >> 6) * WRP + (t & 63)] = qWr[(size_t)bl * 512 + t];
  if (t < HH * SS) qWdS[(t >> 4) * WDP + (t & 15)] = qWd[(size_t)bl * 128 + t];
  if (t < HH) qbS[t] = qb[(size_t)bl * HH + t];
  if (t < DD) qlS[t] = src[(size_t)bl * DD + t];
  __syncthreads();

  const float* rfbase = relF + (size_t)bl * (LL * CC);  // uniform -> SGPR pair
  const int msub = t >> 3;
  const int h = t & 7;

  // ---- prologue: async-stage chunk 0 into buffer 0 ----
  {
    unsigned ldsRf = (unsigned)reinterpret_cast<uintptr_t>(&rfS[0][0]);
#pragma unroll
    for (int rep = 0; rep < 2; ++rep) {
      int idx = t + rep * 512;  // 1024 x b128 = 16KB
      int row = idx >> 4, seg = idx & 15;
      unsigned voff = (unsigned)(row * (CC * 4) + seg * 16);
      unsigned loff = ldsRf + (unsigned)((row * RFP + seg * 4) * 4);
      asm volatile("global_load_async_to_lds_b128 %0, %1, %2"
                   :
                   : "v"(loff), "v"(voff), "s"(rfbase)
                   : "memory");
    }
  }

  for (int ch = 0; ch < 8; ++ch) {
    const int m0 = ch * 64;
    const int cur = ch & 1;

    // ---- issue next chunk's copy into the other buffer (overlaps compute) --
    if (ch < 7) {
      unsigned ldsRf = (unsigned)reinterpret_cast<uintptr_t>(&rfS[cur ^ 1][0]);
#pragma unroll
      for (int rep = 0; rep < 2; ++rep) {
        int idx = t + rep * 512;
        int row = idx >> 4, seg = idx & 15;
        unsigned voff = (unsigned)(((ch + 1) * 64 + row) * (CC * 4) + seg * 16);
        unsigned loff = ldsRf + (unsigned)((row * RFP + seg * 4) * 4);
        asm volatile("global_load_async_to_lds_b128 %0, %1, %2"
                     :
                     : "v"(loff), "v"(voff), "s"(rfbase)
                     : "memory");
      }
    }
    // ---- spherical harmonics for this chunk ----
    if (t < 64) {
      const float* dp = relD + ((size_t)bl * LL + m0 + t) * 3;
      float sh[16];
      sph16(dp[0], dp[1], dp[2], sh);
#pragma unroll
      for (int s = 0; s < SS; ++s) sphS[t * SS + s] = sh[s];
    }
    // current buffer's 2 async ops landed when <=2 (next chunk's) outstanding;
    // async loads complete in order.
    if (ch < 7)
      asm volatile("s_wait_asynccnt 0x2" ::: "memory");
    else
      asm volatile("s_wait_asynccnt 0x0" ::: "memory");
    __syncthreads();  // rf[cur] + sph visible to all waves

    // ---- scores: thread owns (m = m0+msub, head h) ----
    const int m = m0 + msub;
    const float4* qp =
        (const float4*)(src + ((size_t)(b * LL + m)) * DD + h * DHD);
    const float* qlp = qlS + h * DHD;
    float bsum = 0.f;
#pragma unroll
    for (int i = 0; i < 8; ++i) {
      float4 qa = qp[i];
      bsum += qa.x * qlp[4 * i] + qa.y * qlp[4 * i + 1] +
              qa.z * qlp[4 * i + 2] + qa.w * qlp[4 * i + 3];
    }
    const float* rrow = rfS[cur] + msub * RFP;
    const float* wrow = qWrS + h * WRP;
    float rsum = 0.f;
#pragma unroll 8
    for (int c = 0; c < CC; ++c) rsum += wrow[c] * rrow[c];
    const float* srow = sphS + msub * SS;
    const float* wd = qWdS + h * WDP;
    float dsum = 0.f;
#pragma unroll
    for (int s = 0; s < SS; ++s) dsum += wd[s] * srow[s];
    scoresS[h * SCP + m] = bsum * scale + rsum + dsum + qbS[h];

    __syncthreads();  // all reads of rf[cur]/sphS done before they are reused
  }

  // --- softmax over m (one wave per head) ---
  if (wave < HH) {
    float* row = scoresS + wave * SCP;
    float mx = -3.0e38f;
#pragma unroll
    for (int k = 0; k < 16; ++k) mx = fmaxf(mx, row[lane + 32 * k]);
    mx = wred_max(mx);
    float sum = 0.f;
#pragma unroll
    for (int k = 0; k < 16; ++k) {
      float e = __expf(row[lane + 32 * k] - mx);
      row[lane + 32 * k] = e;
      sum += e;
    }
    sum = wred_sum(sum);
    float inv = 1.0f / sum;
#pragma unroll
    for (int k = 0; k < 16; ++k) row[lane + 32 * k] *= inv;
  }
  __syncthreads();

  // --- AV + residual + LayerNorm1 (threads 0..255 own channel t) ---
  float x = 0.f;
  if (t < DD) {
    const float* qcol = src + (size_t)b * LL * DD + t;
    const float* arow = scoresS + (t >> 5) * SCP;  // head = t/32
    float o = 0.f;
#pragma unroll 4
    for (int mm = 0; mm < LL; ++mm) o += arow[mm] * qcol[(size_t)mm * DD];
    x = src[(size_t)bl * DD + t] + o;
  }
  if (t < DD) redS[t] = x;
  __syncthreads();
  for (int s2 = 128; s2 > 0; s2 >>= 1) {
    if (t < s2) redS[t] += redS[t + s2];
    __syncthreads();
  }
  float mean = redS[0] * (1.0f / DD);
  __syncthreads();
  if (t < DD) {
    float dxy = x - mean;
    redS[t] = dxy * dxy;
  }
  __syncthreads();
  for (int s2 = 128; s2 > 0; s2 >>= 1) {
    if (t < s2) redS[t] += redS[t + s2];
    __syncthreads();
  }
  float var = redS[0] * (1.0f / DD);
  if (t < DD) {
    float xn = (x - mean) * rsqrtf(var + 1e-5f);
    x1out[(size_t)bl * DD + t] = xn * g1[t] + bb1[t];
  }
}

// ============ kernel 3: FFN1 (x1 @ W1 + b1, leaky relu) via WMMA ======
// grid = (8, 128), block 256 (8 waves); wave -> one 16x16 tile
__global__ void __launch_bounds__(256) kffn1(const float* __restrict__ x1,
                                             const float* __restrict__ W1,
                                             const float* __restrict__ b1,
                                             float* __restrict__ h1) {
  const int t = threadIdx.x;
  const int lane = t & 31;
  const int half = lane >> 4;
  const int lr = lane & 15;
  const int n0 = (blockIdx.x * 8 + (t >> 5)) * 16;
  const int m0 = blockIdx.y * 16;

  v8f acc = {0.f, 0.f, 0.f, 0.f, 0.f, 0.f, 0.f, 0.f};
  const float* ap = x1 + (size_t)(m0 + lr) * DD;
  for (int k = 0; k < DD; k += 4) {
    v2f a = *(const v2f*)(ap + k + half * 2);
    v2f bv;
    bv[0] = W1[(size_t)(k + half * 2) * FF + n0 + lr];
    bv[1] = W1[(size_t)(k + half * 2 + 1) * FF + n0 + lr];
    acc = wmma_f32_4(a, bv, acc);
  }
  const int col = n0 + lr;
  const float bias = b1[col];
#pragma unroll
  for (int r = 0; r < 8; ++r) {
    int row = m0 + half * 8 + r;
    float v = acc[r] + bias;
    v = v > 0.f ? v : 0.01f * v;
    h1[(size_t)row * FF + col] = v;
  }
}

// ===== kernel 4: FFN2 (h1 @ W2 + b2) + residual + LayerNorm2 via WMMA ====
// grid = 128 blocks (16-row strips), block 512 (16 waves, one 16x16 tile each)
__global__ void __launch_bounds__(512) kffn2(
    const float* __restrict__ h1, const float* __restrict__ W2,
    const float* __restrict__ b2, const float* __restrict__ x1,
    const float* __restrict__ g2, const float* __restrict__ bb2,
    float* __restrict__ out) {
  __shared__ float xb[16 * DD];
  const int t = threadIdx.x;
  const int wave = t >> 5;
  const int lane = t & 31;
  const int half = lane >> 4;
  const int lr = lane & 15;
  const int m0 = blockIdx.x * 16;
  const int n0 = wave * 16;

  v8f acc = {0.f, 0.f, 0.f, 0.f, 0.f, 0.f, 0.f, 0.f};
  const float* ap = h1 + (size_t)(m0 + lr) * FF;
  for (int k = 0; k < FF; k += 4) {
    v2f a = *(const v2f*)(ap + k + half * 2);
    v2f bv;
    bv[0] = W2[(size_t)(k + half * 2) * DD + n0 + lr];
    bv[1] = W2[(size_t)(k + half * 2 + 1) * DD + n0 + lr];
    acc = wmma_f32_4(a, bv, acc);
  }
  const int col = n0 + lr;
  const float bias = b2[col];
#pragma unroll
  for (int r = 0; r < 8; ++r) {
    int rowl = half * 8 + r;
    xb[rowl * DD + col] = acc[r] + bias + x1[(size_t)(m0 + rowl) * DD + col];
  }
  __syncthreads();

  // LayerNorm per row: wave w handles row w
  {
    const float* row = xb + wave * DD;
    float s = 0.f;
#pragma unroll
    for (int k = 0; k < 8; ++k) s += row[lane + 32 * k];
    s = wred_sum(s);
    float mean = s * (1.0f / DD);
    float v = 0.f;
#pragma unroll
    for (int k = 0; k < 8; ++k) {
      float d = row[lane + 32 * k] - mean;
      v += d * d;
    }
    v = wred_sum(v);
    float rinv = rsqrtf(v * (1.0f / DD) + 1e-5f);
    float* orow = out + (size_t)(m0 + wave) * DD;
#pragma unroll
    for (int k = 0; k < 8; ++k) {
      int c = lane + 32 * k;
      orow[c] = (row[c] - mean) * rinv * g2[c] + bb2[c];
    }
  }
}

// ---------------------------------------------------------------------
extern "C" void kernel_launch(void* const* d_in, const int* in_sizes, int n_in,
                              void* d_out, int out_size, void* d_ws,
                              size_t ws_size, hipStream_t stream) {
  (void)in_sizes; (void)n_in; (void)out_size; (void)ws_size;
  const float* src = (const float*)d_in[0];
  const float* relF = (const float*)d_in[1];
  const float* relD = (const float*)d_in[2];
  const float* relW = (const float*)d_in[4];
  const float* relb = (const float*)d_in[5];
  const float* dirW = (const float*)d_in[6];
  const float* dirb = (const float*)d_in[7];
  const float* W1 = (const float*)d_in[8];
  const float* b1 = (const float*)d_in[9];
  const float* W2 = (const float*)d_in[10];
  const float* b2 = (const float*)d_in[11];
  const float* g1 = (const float*)d_in[12];
  const float* bb1 = (const float*)d_in[13];
  const float* g2 = (const float*)d_in[14];
  const float* bb2 = (const float*)d_in[15];
  float* out = (float*)d_out;
  float* ws = (float*)d_ws;

  float* qWr = ws;                  // 4*512*8*64   = 1048576 floats
  float* qWd = ws + 1048576;        // 4*512*8*16   = 262144
  float* qb = ws + 1310720;         // 4*512*8      = 16384
  float* x1 = ws + 1327104;         // 4*512*256    = 524288
  float* h1 = ws + 1851392;         // 4*512*1024   = 2097152  (end ~15.8 MB)

  kprep<<<BB * LL, 256, 0, stream>>>(src, relW, relb, dirW, dirb, qWr, qWd, qb);
  kattn<<<BB * LL, 512, 0, stream>>>(src, relF, relD, qWr, qWd, qb, g1, bb1,
                                     x1);
  kffn1<<<dim3(8, 128), 256, 0, stream>>>(x1, W1, b1, h1);
  kffn2<<<128, 512, 0, stream>>>(h1, W2, b2, x1, g2, bb2, out);
}